// ImprovedGCN_52295521796229
// MI455X (gfx1250) — compile-verified
//
#include <hip/hip_runtime.h>
#include <hip/hip_bf16.h>

typedef __attribute__((ext_vector_type(2))) float v2f;
typedef __attribute__((ext_vector_type(8))) float v8f;

// ---------------------------------------------------------------------------
// GEMM via V_WMMA_F32_16X16X4_F32: C[M,N] = A[M,K] @ B[K,N] (+ bias[N])
// One wave computes one 16x16 tile of C, accumulating over K in steps of 4.
//
// VGPR layouts (CDNA5 ISA 7.12.2, wave32):
//   A (16x4 f32, 2 VGPRs): lanes 0-15 -> rows M=0..15, vgpr0=K0 vgpr1=K1;
//                          lanes 16-31 -> same rows,   vgpr0=K2 vgpr1=K3.
//   B (4x16 f32, 2 VGPRs): lanes 0-15 -> cols N=0..15, vgpr0=K0 vgpr1=K1;
//                          lanes 16-31 -> same cols,   vgpr0=K2 vgpr1=K3.
//   C/D (16x16 f32, 8 VGPRs): vgpr r: lanes 0-15 -> row M=r, lanes 16-31 -> M=r+8.
// ---------------------------------------------------------------------------
__global__ void gcn_gemm_wmma(const float* __restrict__ A,
                              const float* __restrict__ B,
                              const float* __restrict__ bias,
                              float* __restrict__ C,
                              int M, int N, int K) {
  const int wave = (int)((blockIdx.x * (unsigned)blockDim.x + threadIdx.x) >> 5);
  const int lane = (int)(threadIdx.x & 31u);
  const int ntn  = N >> 4;                 // # of 16-wide tiles along N
  const int mt   = wave / ntn;
  const int nt   = wave - mt * ntn;
  if (mt * 16 >= M) return;                // whole-wave exit: EXEC stays all-1s

  const int half = lane >> 4;              // 0: K0/K1 slice, 1: K2/K3 slice
  const int l16  = lane & 15;

  // 32-bit offsets: max product here is 50000*128 < 2^31.
  const float* arow = A + (mt * 16 + l16) * K;
  const float* bcol = B + (nt * 16 + l16);

  v8f acc = {0.f, 0.f, 0.f, 0.f, 0.f, 0.f, 0.f, 0.f};

  for (int k = 0; k < K; k += 4) {
    const int ka = k + half * 2;
    v2f a, b;
    a.x = arow[ka];
    a.y = arow[ka + 1];
    b.x = bcol[ka * N];
    b.y = bcol[(ka + 1) * N];
    // (neg_a, A, neg_b, B, c_mod, C, reuse_a, reuse_b)
    acc = __builtin_amdgcn_wmma_f32_16x16x4_f32(false, a, false, b,
                                                (short)0, acc, false, false);
  }

  const float bv = bias ? bias[nt * 16 + l16] : 0.0f;
  float* crow = C + (mt * 16 + half * 8) * N + (nt * 16 + l16);
#pragma unroll
  for (int r = 0; r < 8; ++r) {
    crow[r * N] = acc[r] + bv;
  }
}

// ---------------------------------------------------------------------------
// Native fp32 global atomic add (no return). Guarantees GLOBAL_ATOMIC_ADD_F32
// rather than a CAS loop. S_ENDPGM performs an implicit wait-idle, and the
// memory clobber orders it against surrounding accesses.
// ---------------------------------------------------------------------------
__device__ __forceinline__ void atomic_add_f32(float* p, float v) {
  asm volatile("global_atomic_add_f32 %0, %1, off"
               :
               : "v"(p), "v"(v)
               : "memory");
}

// ---------------------------------------------------------------------------
// Edge scatter: agg[dst[e], :] += w[e] * xn[src[e], :]
// One lane per (edge, float4 chunk); chunk count is a power of two passed as
// log2 so the edge/chunk split is shift+mask. For D=128 one wave covers one
// 512B src row (fully coalesced gather).
// ---------------------------------------------------------------------------
__global__ void gcn_edge_scatter(const float* __restrict__ xn,
                                 const int* __restrict__ src,
                                 const int* __restrict__ dst,
                                 const float* __restrict__ w,
                                 float* __restrict__ agg,
                                 int E, int D, int lg2chunks) {
  const long long tid   = (long long)blockIdx.x * blockDim.x + threadIdx.x;
  const long long total = (long long)E << lg2chunks;
  if (tid >= total) return;

  const int e  = (int)(tid >> lg2chunks);
  const int c4 = ((int)tid & ((1 << lg2chunks) - 1)) << 2;

  const int   s  = src[e];
  const int   d  = dst[e];
  const float we = w[e];

  const float4 v = *(const float4*)(xn + s * D + c4);   // s*D < 2^31
  float* out = agg + d * D + c4;
  atomic_add_f32(out + 0, we * v.x);
  atomic_add_f32(out + 1, we * v.y);
  atomic_add_f32(out + 2, we * v.z);
  atomic_add_f32(out + 3, we * v.w);
}

// ---------------------------------------------------------------------------
// Elementwise tanh over n4 float4 chunks.
// ---------------------------------------------------------------------------
__global__ void gcn_tanh(float* __restrict__ h, int n4) {
  const int tid = (int)(blockIdx.x * blockDim.x + threadIdx.x);
  if (tid >= n4) return;
  float4 v = *((float4*)h + tid);
  v.x = tanhf(v.x); v.y = tanhf(v.y); v.z = tanhf(v.z); v.w = tanhf(v.w);
  *((float4*)h + tid) = v;
}

// ---------------------------------------------------------------------------
// Host-side orchestration
// ---------------------------------------------------------------------------
static inline void launch_gemm(const float* A, const float* B, const float* bias,
                               float* C, int M, int N, int K, hipStream_t s) {
  const int waves   = ((M + 15) / 16) * (N / 16);
  const int threads = 256;                          // 8 waves / block
  const int blocks  = (waves * 32 + threads - 1) / threads;
  gcn_gemm_wmma<<<blocks, threads, 0, s>>>(A, B, bias, C, M, N, K);
}

static inline void launch_scatter(const float* xn, const int* src, const int* dst,
                                  const float* w, float* agg, int E, int D,
                                  int lg2chunks, hipStream_t s) {
  const long long total = (long long)E << lg2chunks;
  const int threads = 256;
  const int blocks  = (int)((total + threads - 1) / threads);
  gcn_edge_scatter<<<blocks, threads, 0, s>>>(xn, src, dst, w, agg, E, D, lg2chunks);
}

extern "C" void kernel_launch(void* const* d_in, const int* in_sizes, int n_in,
                              void* d_out, int out_size, void* d_ws, size_t ws_size,
                              hipStream_t stream) {
  constexpr int N_NODES = 50000;
  constexpr int DIN = 128, DH = 128, DOUT = 64;

  const float* x          = (const float*)d_in[0];
  const int*   edge_index = (const int*)d_in[1];
  const float* edge_w     = (const float*)d_in[2];
  const float* W_nbr1     = (const float*)d_in[3];
  const float* W_own1     = (const float*)d_in[4];
  const float* b1         = (const float*)d_in[5];
  const float* W_nbr2     = (const float*)d_in[6];
  const float* W_own2     = (const float*)d_in[7];
  const float* b2         = (const float*)d_in[8];

  const int E = in_sizes[1] / 2;            // edge_index is [2, E]
  const int* src = edge_index;              // row 0
  const int* dst = edge_index + E;          // row 1

  float* out = (float*)d_out;

  // Workspace carving (floats): xn[50000*128] | h[50000*128] | hn[50000*64]
  float* xn = (float*)d_ws;
  float* h  = xn + (size_t)N_NODES * DH;
  float* hn = h  + (size_t)N_NODES * DH;
  (void)ws_size; (void)n_in; (void)out_size;

  // ---- Layer 1 ----
  launch_gemm(x, W_nbr1, nullptr, xn, N_NODES, DH, DIN, stream);   // xn = x @ W_nbr1
  launch_gemm(x, W_own1, b1,      h,  N_NODES, DH, DIN, stream);   // h  = x @ W_own1 + b1
  launch_scatter(xn, src, dst, edge_w, h, E, DH, /*lg2(128/4)=*/5, stream);
  {
    const int n4 = N_NODES * DH / 4;
    gcn_tanh<<<(n4 + 255) / 256, 256, 0, stream>>>(h, n4);         // h = tanh(h)
  }

  // ---- Layer 2 ----
  launch_gemm(h, W_nbr2, nullptr, hn,  N_NODES, DOUT, DH, stream); // hn  = h @ W_nbr2
  launch_gemm(h, W_own2, b2,      out, N_NODES, DOUT, DH, stream); // out = h @ W_own2 + b2
  launch_scatter(hn, src, dst, edge_w, out, E, DOUT, /*lg2(64/4)=*/4, stream);
}